// LayeredShader_55765855371714
// MI455X (gfx1250) — compile-verified
//
#include <hip/hip_runtime.h>
#include <stdint.h>

// Layered RGB blend (pytorch3d layered_rgb_blend), N=16 H=256 W=256 K=8 C=3.
// Memory-bound (0.34 FLOP/byte): optimized as an async-DMA streaming kernel
// for CDNA5 (gfx1250) using GLOBAL_LOAD_ASYNC_TO_LDS_B128 + ASYNCcnt
// double buffering, NT cache hints, conflict-free SoA LDS layout.

#define K_FRAGS        8
#define SIGMA_INV      10000.0f      // 1 / 1e-4
#define PIX_PER_TILE   256
#define TILES_PER_BLK  4
#define HW_SHIFT       16            // H*W = 65536
#define HW_MASK        65535

#define COLORS_TILE_B  (PIX_PER_TILE * 96)   // 24576 B
#define DISTS_TILE_B   (PIX_PER_TILE * 32)   // 8192 B
#define BUF_B          (COLORS_TILE_B + 2 * DISTS_TILE_B)  // 40960 B
#define LDS_DIST_OFF   COLORS_TILE_B
#define LDS_PIX_OFF    (COLORS_TILE_B + DISTS_TILE_B)

// One 16-byte async DMA transfer: global -> LDS, non-temporal (read-once stream).
__device__ __forceinline__ void async_copy_b128(uint32_t lds_byte_addr,
                                                const void* gptr) {
    asm volatile("global_load_async_to_lds_b128 %0, %1, off th:TH_LOAD_NT"
                 :: "v"(lds_byte_addr), "v"(gptr)
                 : "memory");
}

__device__ __forceinline__ void issue_tile(uint32_t lds_base,
                                           const char* cbase,
                                           const char* dbase,
                                           const char* pbase,
                                           int tid) {
    // colors: pixel tid owns bytes [96*tid, 96*tid+96); SoA in LDS: chunk j at
    // lds_base + j*4096 + tid*16  -> compute-side reads are lane-stride 16B.
    #pragma unroll
    for (int j = 0; j < 6; ++j)
        async_copy_b128(lds_base + j * 4096 + tid * 16,
                        cbase + tid * 96 + j * 16);
    #pragma unroll
    for (int j = 0; j < 2; ++j)
        async_copy_b128(lds_base + LDS_DIST_OFF + j * 4096 + tid * 16,
                        dbase + tid * 32 + j * 16);
    #pragma unroll
    for (int j = 0; j < 2; ++j)
        async_copy_b128(lds_base + LDS_PIX_OFF + j * 4096 + tid * 16,
                        pbase + tid * 32 + j * 16);
}

__global__ __launch_bounds__(PIX_PER_TILE)
void layered_blend_kernel(const float* __restrict__ colors,
                          const float* __restrict__ dists,
                          const int*   __restrict__ pix,
                          const float* __restrict__ background,
                          float*       __restrict__ out,
                          int numTiles) {
    __shared__ __align__(16) char smem[2 * BUF_B];

    const int tid = threadIdx.x;
    int t0 = blockIdx.x * TILES_PER_BLK;
    int t1 = t0 + TILES_PER_BLK;
    if (t1 > numTiles) t1 = numTiles;
    if (t0 >= t1) return;

    const float bg0 = background[0];
    const float bg1 = background[1];
    const float bg2 = background[2];

    // Low 32 bits of the generic pointer == workgroup-relative LDS byte address.
    const uint32_t lds0 = (uint32_t)(size_t)(&smem[0]);
    const char* C = (const char*)colors;
    const char* D = (const char*)dists;
    const char* P = (const char*)pix;

    // Prologue: start DMA of first tile into buffer 0.
    issue_tile(lds0,
               C + (size_t)t0 * COLORS_TILE_B,
               D + (size_t)t0 * DISTS_TILE_B,
               P + (size_t)t0 * DISTS_TILE_B, tid);

    int cur = 0;
    for (int tile = t0; tile < t1; ++tile) {
        const bool more = (tile + 1 < t1);
        if (more) {
            // Kick next tile's DMA into the other buffer, then wait only for
            // the current tile (10 newest async ops belong to the next tile;
            // async loads complete in order).
            issue_tile(lds0 + (uint32_t)((cur ^ 1) * BUF_B),
                       C + (size_t)(tile + 1) * COLORS_TILE_B,
                       D + (size_t)(tile + 1) * DISTS_TILE_B,
                       P + (size_t)(tile + 1) * DISTS_TILE_B, tid);
            asm volatile("s_wait_asynccnt 10" ::: "memory");
        } else {
            asm volatile("s_wait_asynccnt 0" ::: "memory");
        }

        const char* sb = smem + cur * BUF_B;

        float col[24];
        #pragma unroll
        for (int j = 0; j < 6; ++j)
            *reinterpret_cast<float4*>(&col[4 * j]) =
                *reinterpret_cast<const float4*>(sb + j * 4096 + tid * 16);

        const float4 dv0 = *reinterpret_cast<const float4*>(sb + LDS_DIST_OFF + tid * 16);
        const float4 dv1 = *reinterpret_cast<const float4*>(sb + LDS_DIST_OFF + 4096 + tid * 16);
        const int4  pv0 = *reinterpret_cast<const int4*>(sb + LDS_PIX_OFF + tid * 16);
        const int4  pv1 = *reinterpret_cast<const int4*>(sb + LDS_PIX_OFF + 4096 + tid * 16);

        const float dk[K_FRAGS] = {dv0.x, dv0.y, dv0.z, dv0.w,
                                   dv1.x, dv1.y, dv1.z, dv1.w};
        const int   pk[K_FRAGS] = {pv0.x, pv0.y, pv0.z, pv0.w,
                                   pv1.x, pv1.y, pv1.z, pv1.w};

        // Front-to-back compositing:
        //   w_k = prod_{j<k}(1-a_j) * a_k ;  a_k = exp(-max(d,0)/sigma)*[face>=0]
        float occ = 1.0f, r = 0.0f, g = 0.0f, b = 0.0f;
        #pragma unroll
        for (int k = 0; k < K_FRAGS; ++k) {
            const float e = __expf(-fmaxf(dk[k], 0.0f) * SIGMA_INV);
            const float a = (pk[k] >= 0) ? e : 0.0f;
            const float w = occ * a;
            r = fmaf(w, col[3 * k + 0], r);
            g = fmaf(w, col[3 * k + 1], g);
            b = fmaf(w, col[3 * k + 2], b);
            occ *= (1.0f - a);
        }
        r = fmaf(occ, bg0, r);
        g = fmaf(occ, bg1, g);
        b = fmaf(occ, bg2, b);

        // Output is [N, C+1, H, W]; p = n*HW + hw with HW = 65536.
        const int p  = tile * PIX_PER_TILE + tid;
        const int n  = p >> HW_SHIFT;
        const int hw = p & HW_MASK;
        const size_t ob = ((size_t)n * 4) << HW_SHIFT;
        out[ob + hw]                         = r;
        out[ob + (1u << HW_SHIFT) + hw]      = g;
        out[ob + (2u << HW_SHIFT) + hw]      = b;
        out[ob + (3u << HW_SHIFT) + hw]      = 1.0f - occ;

        cur ^= 1;
    }
}

extern "C" void kernel_launch(void* const* d_in, const int* in_sizes, int n_in,
                              void* d_out, int out_size, void* d_ws, size_t ws_size,
                              hipStream_t stream) {
    (void)n_in; (void)out_size; (void)d_ws; (void)ws_size;
    const float* colors     = (const float*)d_in[0];
    const float* dists      = (const float*)d_in[1];
    const int*   pix        = (const int*)d_in[2];   // int32 on device
    const float* background = (const float*)d_in[3];
    float* out = (float*)d_out;

    const int totalPix = in_sizes[1] / K_FRAGS;          // N*H*W = 1,048,576
    const int numTiles = totalPix / PIX_PER_TILE;        // 4096
    const int blocks   = (numTiles + TILES_PER_BLK - 1) / TILES_PER_BLK;  // 1024

    hipLaunchKernelGGL(layered_blend_kernel, dim3(blocks), dim3(PIX_PER_TILE),
                       0, stream, colors, dists, pix, background, out, numTiles);
}